// Conv5d_36206574306083
// MI455X (gfx1250) — compile-verified
//
#include <hip/hip_runtime.h>

// ---- problem geometry -------------------------------------------------------
// x   : (B=2, C=4, CD=8, T=8, D=8, H=96, W=96) fp32
// W   : (9, O=4, C=4, 3,3,3) fp32       b : (9,4) fp32
// out : (B=2, O=4, CC=6, TT=6, D=8, H=96, W=96) fp32
constexpr int KTOT = 9 * 4 * 3 * 6 * 3;     // 1944 expanded-K (kh' in 0..5)
constexpr int NQ   = KTOT / 4;              // 486 wmma_f32_16x16x4 per tile
constexpr int WLDS_FLOATS = KTOT * 16;      // 31104 floats, layout [k][m]
constexpr int TAB_INTS    = KTOT * 2;       // {byte_off, meta} per k
constexpr int SMEM_BYTES  = (WLDS_FLOATS + TAB_INTS) * 4;   // 139,968 B

// x strides (elements)
constexpr int XS_H = 96, XS_D = 9216, XS_T = 73728, XS_CD = 589824,
              XS_C = 4718592, XS_B = 18874368;
// out strides (elements)
constexpr int OS_H = 96, OS_D = 9216, OS_TT = 73728, OS_CC = 442368,
              OS_O = 2654208, OS_B = 10616832;

typedef float v2f __attribute__((ext_vector_type(2)));
typedef float v8f __attribute__((ext_vector_type(8)));

__device__ __forceinline__ void decode_k(int k, int& win, int& c, int& kd,
                                         int& khp, int& kw) {
  kw = k % 3;
  int r3 = k / 3;
  khp = r3 % 6;              // expanded kh' in 0..5
  int blk = r3 / 6;          // (win,c,kd), 108 blocks
  kd  = blk % 3;
  c   = (blk / 3) & 3;
  win = blk / 12;            // i*3+j, 0..8
}

__global__ __launch_bounds__(256, 2)
void conv5d_wmma(const float* __restrict__ x, const float* __restrict__ Wt,
                 const float* __restrict__ bias, float* __restrict__ out) {
  extern __shared__ float smem[];
  float* wlds = smem;                          // W' expanded weights [k][16]
  int*   tab  = (int*)(smem + WLDS_FLOATS);    // per-k {byte_off, meta}

  const int tid  = threadIdx.x;
  const int lane = tid & 31;
  const int wid  = tid >> 5;        // 8 waves
  const int hi   = lane >> 4;       // lane half (A: K pair select, B: K row pair)
  const int mn   = lane & 15;       // m for A rows, n for B and C/D columns

  // block -> (b, cc, tt, d, h-group of 16 rows)
  int bid = blockIdx.x;
  const int hg = bid % 6;  bid /= 6;
  const int d  = bid & 7;  bid >>= 3;
  const int tt = bid % 6;  bid /= 6;
  const int cc = bid % 6;  bid /= 6;
  const int b  = bid;

  // ---- stage expanded weights W'[k][m], zeros where kh'-dh outside [0,3) ----
  for (int idx = tid; idx < WLDS_FLOATS; idx += 256) {
    int k = idx >> 4, m = idx & 15;
    int win, c, kd, khp, kw;
    decode_k(k, win, c, kd, khp, kw);
    int o = m >> 2, dh = m & 3;
    int kh = khp - dh;
    float v = 0.0f;
    if (kh >= 0 && kh < 3)
      v = Wt[((win * 4 + o) * 4 + c) * 27 + kd * 9 + kh * 3 + kw] * (1.0f / 9.0f);
    wlds[k * 16 + m] = v;
  }

  // ---- stage per-k relative byte offset + padding metadata ----
  for (int k = tid; k < KTOT; k += 256) {
    int win, c, kd, khp, kw;
    decode_k(k, win, c, kd, khp, kw);
    int i = win / 3, j = win % 3;
    int off = c * XS_C + i * XS_CD + j * XS_T + (kd - 1) * XS_D +
              (khp - 1) * XS_H + (kw - 1);
    tab[k * 2 + 0] = off * 4;                       // bytes
    tab[k * 2 + 1] = kd | (khp << 8) | (kw << 16);  // biased deltas (+1)
  }

  // ---- mean bias (uniform -> scalar loads) ----
  float mb[4];
#pragma unroll
  for (int o = 0; o < 4; ++o) {
    float s = 0.0f;
#pragma unroll
    for (int wn = 0; wn < 9; ++wn) s += bias[wn * 4 + o];
    mb[o] = s * (1.0f / 9.0f);
  }
  __syncthreads();

  const float* xb0 = x + b * XS_B + cc * XS_CD + tt * XS_T + d * XS_D;
  const int dm1 = d - 1;

  // 8 waves x 3 tiles cover 4 h-tiles(4 rows) x 6 w-tiles(16 cols) = 16h x 96w
  for (int tsel = 0; tsel < 3; ++tsel) {
    const int tl = wid + tsel * 8;          // 0..23
    const int ht = tl / 6, wt = tl % 6;
    const int h0 = hg * 16 + ht * 4;
    const int w0 = wt * 16;

    const char* xbase = (const char*)(xb0 + h0 * XS_H + w0 + mn);
    const int hm1  = h0 - 1;
    const int wnm1 = w0 + mn - 1;

    v8f acc;
#pragma unroll
    for (int v = 0; v < 8; ++v)             // C/D: M = v + 8*hi, o = M>>2
      acc[v] = hi ? mb[(v >> 2) + 2] : mb[v >> 2];

    for (int q = 0; q < NQ; ++q) {
      const int r0 = q * 4 + 2 * hi;        // this lane's B row pair / A K pair
      const int4 te = *(const int4*)(&tab[r0 * 2]);   // ds_load_b128

      // B row r0
      int md = te.y;
      bool ok0 = ((unsigned)(dm1 + (md & 0xff)) < 8u) &
                 ((unsigned)(hm1 + ((md >> 8) & 0xff)) < 96u) &
                 ((unsigned)(wnm1 + ((md >> 16) & 0xff)) < 96u);
      int  o0 = ok0 ? te.x : 0;
      float b0 = *(const float*)(xbase + o0);
      b0 = ok0 ? b0 : 0.0f;

      // B row r0+1
      md = te.w;
      bool ok1 = ((unsigned)(dm1 + (md & 0xff)) < 8u) &
                 ((unsigned)(hm1 + ((md >> 8) & 0xff)) < 96u) &
                 ((unsigned)(wnm1 + ((md >> 16) & 0xff)) < 96u);
      int  o1 = ok1 ? te.z : 0;
      float b1 = *(const float*)(xbase + o1);
      b1 = ok1 ? b1 : 0.0f;

      // A 16x4: lane m = mn, VGPR0 = A[m, k0+2*hi], VGPR1 = A[m, k0+1+2*hi]
      v2f Av; Av.x = wlds[r0 * 16 + mn];
              Av.y = wlds[r0 * 16 + 16 + mn];
      v2f Bv; Bv.x = b0; Bv.y = b1;

      acc = __builtin_amdgcn_wmma_f32_16x16x4_f32(
          /*neg_a=*/false, Av, /*neg_b=*/false, Bv,
          /*c_mod=*/0, acc, /*reuse_a=*/false, /*reuse_b=*/false);
    }

    // store: D[m=v+8*hi, n] -> out[o=(v>>2)+2*hi, h0+(v&3), w0+n]
    const long ob = (long)b * OS_B + (long)cc * OS_CC + (long)tt * OS_TT +
                    (long)d * OS_D + (long)h0 * OS_H + w0 + mn;
#pragma unroll
    for (int v = 0; v < 8; ++v) {
      int o  = (v >> 2) + 2 * hi;
      int dh = v & 3;
      out[ob + (long)o * OS_O + (long)dh * OS_H] = acc[v];
    }
  }
}

extern "C" void kernel_launch(void* const* d_in, const int* in_sizes, int n_in,
                              void* d_out, int out_size, void* d_ws, size_t ws_size,
                              hipStream_t stream) {
  (void)in_sizes; (void)n_in; (void)out_size; (void)d_ws; (void)ws_size;
  const float* x  = (const float*)d_in[0];
  const float* Wt = (const float*)d_in[1];
  const float* bs = (const float*)d_in[2];
  float* out = (float*)d_out;

  hipFuncSetAttribute((const void*)conv5d_wmma,
                      hipFuncAttributeMaxDynamicSharedMemorySize, SMEM_BYTES);

  // grid = (((b*6+cc)*6+tt)*8+d)*6 + hg  -> 2*6*6*8*6 = 3456 blocks
  conv5d_wmma<<<dim3(3456), dim3(256), SMEM_BYTES, stream>>>(x, Wt, bs, out);
}